// Grid_attention_linear_29231547416585
// MI455X (gfx1250) — compile-verified
//
#include <hip/hip_runtime.h>
#include <hip/hip_bf16.h>

typedef __attribute__((ext_vector_type(2))) float v2f;
typedef __attribute__((ext_vector_type(8))) float v8f;

// ---------------------------------------------------------------------------
// Kernel 1: per-batch GEMM  gc[b, pix, ch] = W[64,128] x g[b,128,4096] + bias
// (conv commuted before upsample). One wave per 16(ch) x 16(pix) tile.
// WMMA f32 16x16x4, K=128 -> 32 steps.
// A layout (ISA 7.12.2, 32-bit A 16x4): lanes 0-15: v0=K0,v1=K1 ; lanes 16-31: v0=K2,v1=K3
// D layout: VGPR v holds M = v + 8*(lane>=16), N = lane%16
// Output stored pixel-major: gcT[b][pix][ch] for contiguous channel reads later.
// ---------------------------------------------------------------------------
__global__ __launch_bounds__(128)
void conv1x1_wmma_kernel(const float* __restrict__ g,    // [8,128,64,64]
                         const float* __restrict__ W,    // [64,128]
                         const float* __restrict__ bias, // [64]
                         float* __restrict__ gcT)        // [8,4096,64]
{
    const int lane = threadIdx.x & 31;
    const int wave = threadIdx.x >> 5;
    const int tile = blockIdx.x * 4 + wave;     // 8192 tiles total

    const int chTile  = tile & 3;               // 4 channel tiles of 16
    const int pixTile = (tile >> 2) & 255;      // 256 pixel tiles of 16
    const int b       = tile >> 10;             // 8 batches

    const int mBase = chTile * 16;
    const int nBase = pixTile * 16;
    const int half  = lane >> 4;                // 0: K lo pair, 1: K hi pair
    const int l15   = lane & 15;

    const float* __restrict__ gB = g + (size_t)b * 128 * 4096;
    const float* __restrict__ Wr = W + (size_t)(mBase + l15) * 128;

    v8f acc = {};
    for (int k0 = 0; k0 < 128; k0 += 4) {
        const int k = k0 + 2 * half;
        v2f a;
        a.x = Wr[k + 0];
        a.y = Wr[k + 1];
        v2f bb;
        bb.x = gB[(size_t)(k + 0) * 4096 + nBase + l15];
        bb.y = gB[(size_t)(k + 1) * 4096 + nBase + l15];
        acc = __builtin_amdgcn_wmma_f32_16x16x4_f32(
            /*neg_a=*/false, a, /*neg_b=*/false, bb,
            /*c_mod=*/(short)0, acc, /*reuse_a=*/false, /*reuse_b=*/false);
    }

    float* __restrict__ outB = gcT + ((size_t)b * 4096 + (nBase + l15)) * 64;
#pragma unroll
    for (int v = 0; v < 8; ++v) {
        const int ch = mBase + v + 8 * half;
        outB[ch] = acc[v] + bias[ch];
    }
}

// ---------------------------------------------------------------------------
// Kernel 2: bilinear upsample (x4, align_corners=False semantics, edge-
// renormalized == clamped here) + per-pixel squared channel sum.
// One thread per output pixel; 4 neighbor pixel-vectors of 64 contiguous
// floats each, float4 loads. Writes RAW att into d_out att region.
// ---------------------------------------------------------------------------
__global__ __launch_bounds__(256)
void upsample_sqsum_kernel(const float* __restrict__ gcT,  // [8,4096,64]
                           float* __restrict__ attRaw)     // [8,65536]
{
    const int idx = blockIdx.x * 256 + threadIdx.x;  // b*65536 + oy*256 + ox
    const int b  = idx >> 16;
    const int p  = idx & 65535;
    const int oy = p >> 8;
    const int ox = p & 255;

    const float sy = (oy + 0.5f) * 0.25f - 0.5f;
    const float sx = (ox + 0.5f) * 0.25f - 0.5f;
    int y0 = (int)floorf(sy); const float fy = sy - (float)y0;
    int x0 = (int)floorf(sx); const float fx = sx - (float)x0;
    int y1 = min(63, max(0, y0 + 1)); y0 = min(63, max(0, y0));
    int x1 = min(63, max(0, x0 + 1)); x0 = min(63, max(0, x0));

    const float w00 = (1.f - fy) * (1.f - fx);
    const float w01 = (1.f - fy) * fx;
    const float w10 = fy * (1.f - fx);
    const float w11 = fy * fx;

    const float* __restrict__ base = gcT + (size_t)b * 4096 * 64;
    const float4* __restrict__ p00 = (const float4*)(base + (size_t)(y0 * 64 + x0) * 64);
    const float4* __restrict__ p01 = (const float4*)(base + (size_t)(y0 * 64 + x1) * 64);
    const float4* __restrict__ p10 = (const float4*)(base + (size_t)(y1 * 64 + x0) * 64);
    const float4* __restrict__ p11 = (const float4*)(base + (size_t)(y1 * 64 + x1) * 64);

    float s = 0.f;
#pragma unroll 4
    for (int c4 = 0; c4 < 16; ++c4) {
        const float4 a = p00[c4], bb = p01[c4], c = p10[c4], d = p11[c4];
        float v;
        v = w00 * a.x + w01 * bb.x + w10 * c.x + w11 * d.x; s += v * v;
        v = w00 * a.y + w01 * bb.y + w10 * c.y + w11 * d.y; s += v * v;
        v = w00 * a.z + w01 * bb.z + w10 * c.z + w11 * d.z; s += v * v;
        v = w00 * a.w + w01 * bb.w + w10 * c.w + w11 * d.w; s += v * v;
    }
    attRaw[idx] = s;
}

// ---------------------------------------------------------------------------
// Kernel 3: per-batch min and sum of raw att. sum(att - min) = sum - N*min.
// stats[b*2] = min, stats[b*2+1] = 1/denominator.
// ---------------------------------------------------------------------------
__global__ __launch_bounds__(256)
void batch_stats_kernel(const float* __restrict__ attRaw,  // [8,65536]
                        float* __restrict__ stats)         // [16]
{
    __shared__ float smin[256];
    __shared__ float ssum[256];
    const int b = blockIdx.x;
    const float4* __restrict__ a = (const float4*)(attRaw + (size_t)b * 65536);

    float mn = 3.4028235e38f, sm = 0.f;
    for (int i = threadIdx.x; i < 16384; i += 256) {
        const float4 v = a[i];
        mn = fminf(mn, fminf(fminf(v.x, v.y), fminf(v.z, v.w)));
        sm += v.x + v.y + v.z + v.w;
    }
    smin[threadIdx.x] = mn;
    ssum[threadIdx.x] = sm;
    __syncthreads();
    for (int s = 128; s > 0; s >>= 1) {
        if ((int)threadIdx.x < s) {
            smin[threadIdx.x] = fminf(smin[threadIdx.x], smin[threadIdx.x + s]);
            ssum[threadIdx.x] += ssum[threadIdx.x + s];
        }
        __syncthreads();
    }
    if (threadIdx.x == 0) {
        const float mnv = smin[0];
        stats[b * 2 + 0] = mnv;
        stats[b * 2 + 1] = 1.0f / (ssum[0] - 65536.0f * mnv);
    }
}

// ---------------------------------------------------------------------------
// Kernel 4: normalize att in place: att = (raw - min) * invDenom.
// Safe in-place: kernel 2 rewrites every element on every launch.
// ---------------------------------------------------------------------------
__global__ __launch_bounds__(256)
void normalize_att_kernel(float* __restrict__ att,         // [8,65536] in d_out
                          const float* __restrict__ stats) // [16]
{
    const int idx = blockIdx.x * 256 + threadIdx.x;
    const int b = idx >> 16;
    const float mn  = stats[b * 2 + 0];
    const float inv = stats[b * 2 + 1];
    att[idx] = (att[idx] - mn) * inv;
}

// ---------------------------------------------------------------------------
// Kernel 5: out[b,c] = dot(x[b,c,:], att[b,:]).  Bandwidth-bound (128 MB of
// x). One block per (b,c) row, float4 streaming, LDS tree reduction.
// ---------------------------------------------------------------------------
__global__ __launch_bounds__(256)
void weighted_sum_kernel(const float* __restrict__ x,    // [8,64,65536]
                         const float* __restrict__ att,  // [8,65536] normalized
                         float* __restrict__ out)        // [8,64]
{
    __shared__ float ssum[256];
    const int bc = blockIdx.x;      // b*64 + c, 512 blocks
    const int b  = bc >> 6;
    const float4* __restrict__ xr = (const float4*)(x + (size_t)bc * 65536);
    const float4* __restrict__ ar = (const float4*)(att + (size_t)b * 65536);

    float s = 0.f;
    for (int i = threadIdx.x; i < 16384; i += 256) {
        const float4 xv = xr[i];
        const float4 av = ar[i];
        s += xv.x * av.x + xv.y * av.y + xv.z * av.z + xv.w * av.w;
    }
    ssum[threadIdx.x] = s;
    __syncthreads();
    for (int st = 128; st > 0; st >>= 1) {
        if ((int)threadIdx.x < st) ssum[threadIdx.x] += ssum[threadIdx.x + st];
        __syncthreads();
    }
    if (threadIdx.x == 0) out[bc] = ssum[0];
}

extern "C" void kernel_launch(void* const* d_in, const int* in_sizes, int n_in,
                              void* d_out, int out_size, void* d_ws, size_t ws_size,
                              hipStream_t stream) {
    const float* x    = (const float*)d_in[0];  // [8,64,256,256]
    const float* g    = (const float*)d_in[1];  // [8,128,64,64]
    const float* W    = (const float*)d_in[2];  // [64,128]
    const float* bias = (const float*)d_in[3];  // [64]

    float* out = (float*)d_out;       // [8,64]   (512 floats)
    float* att = out + 512;           // [8,256,256] (524288 floats)

    float* gcT   = (float*)d_ws;                 // 8*4096*64 floats = 8 MB
    float* stats = gcT + (size_t)8 * 4096 * 64;  // 16 floats

    // 1) small conv via WMMA: 8192 tiles, 4 waves (128 thr) per block
    conv1x1_wmma_kernel<<<2048, 128, 0, stream>>>(g, W, bias, gcT);

    // 2) fused upsample + squared channel-sum -> raw att (in d_out att slot)
    upsample_sqsum_kernel<<<2048, 256, 0, stream>>>(gcT, att);

    // 3) per-batch min / sum
    batch_stats_kernel<<<8, 256, 0, stream>>>(att, stats);

    // 4) normalize att in place
    normalize_att_kernel<<<2048, 256, 0, stream>>>(att, stats);

    // 5) out[b,c] = x . att  (dominant: streams 128 MB, ~5.5 us at 23.3 TB/s)
    weighted_sum_kernel<<<512, 256, 0, stream>>>(x, att, out);
}